// GraphDense_42941083025465
// MI455X (gfx1250) — compile-verified
//
#include <hip/hip_runtime.h>
#include <hip/hip_bf16.h>

// GraphDense: h = x @ W  (WMMA f32 16x16x4), then edge gather-scale-scatter
// with global f32 atomics (resolved in 192MB L2 since h/out are ~12.8MB each).

typedef __attribute__((ext_vector_type(2))) float v2f;
typedef __attribute__((ext_vector_type(8))) float v8f;

#define IN_DIM 128
#define OUT_DIM 32

// ---------------------------------------------------------------------------
// Kernel 1: h[m,n] = sum_k x[m,k] * W[k,n] via V_WMMA_F32_16X16X4_F32.
// One wave computes a 16x32 strip of h (two 16x16 N-tiles, shared A fragment).
// A layout (16x4 f32): lanes 0-15: M=lane, K={k0,k0+1}; lanes 16-31: M=lane-16,
// K={k0+2,k0+3}. B layout (4x16): lanes 0-15: N=lane, K={k0,k0+1}; lanes
// 16-31: N=lane-16, K={k0+2,k0+3}. C/D (16x16 f32): VGPR r -> M=r (lanes
// 0-15) / M=r+8 (lanes 16-31), N=lane&15.
// ---------------------------------------------------------------------------
__global__ __launch_bounds__(256) void gd_gemm_wmma(
    const float* __restrict__ x, const float* __restrict__ W,
    float* __restrict__ h, int n_nodes) {
  const int wave = (int)((blockIdx.x * blockDim.x + threadIdx.x) >> 5);
  const int lane = (int)(threadIdx.x & 31);
  const int n_tiles = (n_nodes + 15) >> 4;
  if (wave >= n_tiles) return;  // uniform per wave: full-wave exit, EXEC stays all-1s for active waves

  const int m_base = wave << 4;
  int mrow = m_base + (lane & 15);
  if (mrow >= n_nodes) mrow = n_nodes - 1;  // clamp reads for (non-existent here) tail tile
  const int khalf = (lane >> 4) << 1;       // 0 for lanes 0-15, 2 for lanes 16-31
  const int ncol  = lane & 15;

  v8f acc0 = {};
  v8f acc1 = {};

  const float* __restrict__ arow = x + (size_t)mrow * IN_DIM;

#pragma unroll 4
  for (int k0 = 0; k0 < IN_DIM; k0 += 4) {
    v2f a;
    a.x = arow[k0 + khalf + 0];
    a.y = arow[k0 + khalf + 1];

    const float* __restrict__ bp = W + (size_t)(k0 + khalf) * OUT_DIM;
    v2f b0, b1;
    b0.x = bp[ncol];
    b0.y = bp[OUT_DIM + ncol];
    b1.x = bp[16 + ncol];
    b1.y = bp[OUT_DIM + 16 + ncol];

    acc0 = __builtin_amdgcn_wmma_f32_16x16x4_f32(
        /*neg_a=*/false, a, /*neg_b=*/false, b0,
        /*c_mod=*/(short)0, acc0, /*reuse_a=*/false, /*reuse_b=*/false);
    acc1 = __builtin_amdgcn_wmma_f32_16x16x4_f32(
        /*neg_a=*/false, a, /*neg_b=*/false, b1,
        /*c_mod=*/(short)0, acc1, /*reuse_a=*/false, /*reuse_b=*/false);
  }

  // Store D: row = m_base + r + (lane>=16 ? 8 : 0), col = ntile*16 + (lane&15)
  const int moff = (lane >> 4) << 3;  // 0 or 8
#pragma unroll
  for (int r = 0; r < 8; ++r) {
    const int row = m_base + moff + r;
    if (row < n_nodes) {
      float* __restrict__ hp = h + (size_t)row * OUT_DIM;
      hp[ncol]      = acc0[r];
      hp[16 + ncol] = acc1[r];
    }
  }
}

// ---------------------------------------------------------------------------
// Kernel 2: zero the output (harness poisons d_out before timing).
// ---------------------------------------------------------------------------
__global__ __launch_bounds__(256) void gd_zero(float* __restrict__ out, int n) {
  int i = (int)(blockIdx.x * blockDim.x + threadIdx.x) * 4;
  if (i + 3 < n) {
    *(float4*)(out + i) = make_float4(0.f, 0.f, 0.f, 0.f);
  } else {
    for (int k = i; k < n; ++k) out[k] = 0.f;
  }
}

// ---------------------------------------------------------------------------
// Kernel 3: one wave per edge; lane = output column (OUT_DIM==32==wave32).
// out[row, lane] += val * h[col, lane]  via global_atomic_add_f32.
// h and out are L2-resident (12.8MB each << 192MB L2).
// ---------------------------------------------------------------------------
__global__ __launch_bounds__(256) void gd_edges(
    const int* __restrict__ rows, const int* __restrict__ cols,
    const float* __restrict__ vals, const float* __restrict__ h,
    float* __restrict__ out, int n_edges) {
  const int wave = (int)((blockIdx.x * blockDim.x + threadIdx.x) >> 5);
  const int lane = (int)(threadIdx.x & 31);
  if (wave >= n_edges) return;

  const int   r = rows[wave];
  const int   c = cols[wave];
  const float v = vals[wave];

  const float hv = h[(size_t)c * OUT_DIM + lane];
  atomicAdd(&out[(size_t)r * OUT_DIM + lane], v * hv);
}

extern "C" void kernel_launch(void* const* d_in, const int* in_sizes, int n_in,
                              void* d_out, int out_size, void* d_ws, size_t ws_size,
                              hipStream_t stream) {
  const float* x         = (const float*)d_in[0];
  const float* W         = (const float*)d_in[1];
  const int*   edge_rows = (const int*)d_in[2];
  const int*   edge_cols = (const int*)d_in[3];
  const float* edge_vals = (const float*)d_in[4];
  float*       out       = (float*)d_out;

  const int n_nodes = in_sizes[0] / IN_DIM;  // 100000
  const int n_edges = in_sizes[2];           // 1600000

  float* h = (float*)d_ws;  // n_nodes * OUT_DIM floats = 12.8 MB

  // 1) h = x @ W  (one wave per 16-row tile, 8 waves per block)
  {
    const int n_tiles = (n_nodes + 15) / 16;
    const int blocks  = (n_tiles + 7) / 8;
    gd_gemm_wmma<<<blocks, 256, 0, stream>>>(x, W, h, n_nodes);
  }

  // 2) out = 0
  {
    const int n      = n_nodes * OUT_DIM;
    const int blocks = (n / 4 + 255) / 256;
    gd_zero<<<blocks, 256, 0, stream>>>(out, n);
  }

  // 3) scatter-accumulate edges (one wave per edge)
  {
    const int blocks = (n_edges + 7) / 8;  // 8 waves / 256-thread block
    gd_edges<<<blocks, 256, 0, stream>>>(edge_rows, edge_cols, edge_vals, h, out, n_edges);
  }
}